// SelfAttentionBlock_32976758898911
// MI455X (gfx1250) — compile-verified
//
#include <hip/hip_runtime.h>

// ---------------------------------------------------------------------------
// Grouped local self-attention block (B=4, C=64, H=W=96, g=8, og=ig=8, k=7).
//
// Stage 1: grouped 1x1 conv for K and V via V_WMMA_F32_16X16X4_F32.
//          Two groups (8+8 channels) packed into one 16x16 block-diagonal
//          weight matrix; K=16 covered by chaining four K=4 WMMAs.
//          (Codegen-verified: 24 v_wmma per wave, coalesced loads/stores.)
// Stage 2: LDS-tiled attention. Each 256-thread block owns (b, g, 16x16
//          pixel tile); the 8-channel 22x22 k/v halo tile (~31 KB) is staged
//          into LDS with zeros outside the image (== reference zero-pad
//          semantics), making the 49-tap softmax loops branch-free with
//          immediate-offset ds_loads.
// ---------------------------------------------------------------------------

typedef __attribute__((ext_vector_type(2))) float v2f;
typedef __attribute__((ext_vector_type(8))) float v8f;

#define B_    4
#define C_    64
#define H_    96
#define W_    96
#define G_    8
#define OG_   8
#define IG_   8
#define KW_   7
#define PAD_  3
#define HW_   (H_ * W_)
#define CHW_  (C_ * HW_)

// ------------------------------ Stage 1 ------------------------------------
// HW_/16 = 576 N-tiles per (b, group-pair); split across waves.
#define NCHUNK 192
#define TILES_PER_CHUNK (HW_ / 16 / NCHUNK)   // 576/192 = 3

__device__ __forceinline__ float blockdiag_w(const float* __restrict__ w,
                                             int gpair, int m, int kk) {
  // w: [G][OG][IG] contiguous (8*8*8); block-diag pairing of groups 2gp,2gp+1
  if (m < 8) {
    return (kk < 8) ? w[(2 * gpair) * 64 + m * 8 + kk] : 0.0f;
  }
  return (kk >= 8) ? w[(2 * gpair + 1) * 64 + (m - 8) * 8 + (kk - 8)] : 0.0f;
}

// One wave (32 threads) per block. blockIdx.x = ((b*4)+gpair)*NCHUNK + chunk.
__global__ void kv_proj_wmma_kernel(const float* __restrict__ x,
                                    const float* __restrict__ wk,
                                    const float* __restrict__ wv,
                                    float* __restrict__ kbuf,
                                    float* __restrict__ vbuf) {
  const int wave  = blockIdx.x;
  const int chunk = wave % NCHUNK;
  const int gp    = (wave / NCHUNK) % 4;      // group pair 0..3
  const int b     = wave / (NCHUNK * 4);
  const int lane  = threadIdx.x;              // 0..31 (wave32)
  const int half  = lane >> 4;
  const int l16   = lane & 15;
  const int ch0   = gp * 16;

  // A fragments (ISA 16x4 f32 layout): lane L holds M = L%16,
  // K = 4*s + 2*(L/16) + {0,1} for step s.
  v2f ak[4], av[4];
#pragma unroll
  for (int s = 0; s < 4; ++s) {
    const int kk = 4 * s + 2 * half;
    ak[s].x = blockdiag_w(wk, gp, l16, kk);
    ak[s].y = blockdiag_w(wk, gp, l16, kk + 1);
    av[s].x = blockdiag_w(wv, gp, l16, kk);
    av[s].y = blockdiag_w(wv, gp, l16, kk + 1);
  }

  const float* xb = x    + (size_t)b * CHW_ + (size_t)ch0 * HW_;
  float*       kb = kbuf + (size_t)b * CHW_ + (size_t)ch0 * HW_;
  float*       vb = vbuf + (size_t)b * CHW_ + (size_t)ch0 * HW_;

  for (int t = 0; t < TILES_PER_CHUNK; ++t) {
    const int n0 = (chunk * TILES_PER_CHUNK + t) * 16;
    v8f acck = {};
    v8f accv = {};
#pragma unroll
    for (int s = 0; s < 4; ++s) {
      // B fragment (4x16 f32): lane L holds rows 4*s + 2*(L/16) + {0,1},
      // column n0 + L%16.
      const int r = 4 * s + 2 * half;
      v2f bfrag;
      bfrag.x = xb[(size_t)r       * HW_ + n0 + l16];
      bfrag.y = xb[(size_t)(r + 1) * HW_ + n0 + l16];
      acck = __builtin_amdgcn_wmma_f32_16x16x4_f32(
          false, ak[s], false, bfrag, (short)0, acck, false, false);
      accv = __builtin_amdgcn_wmma_f32_16x16x4_f32(
          false, av[s], false, bfrag, (short)0, accv, false, false);
    }
    // D layout: VGPR r holds row r + 8*(L/16), column n0 + L%16.
#pragma unroll
    for (int r = 0; r < 8; ++r) {
      const int m = r + 8 * half;
      kb[(size_t)m * HW_ + n0 + l16] = acck[r];
      vb[(size_t)m * HW_ + n0 + l16] = accv[r];
    }
  }
}

// ------------------------------ Stage 2 ------------------------------------
#define TS 16                        // pixel tile edge
#define HALO (TS + 2 * PAD_)         // 22
#define TILE_ELEMS (HALO * HALO)     // 484 per channel

// 256 threads = 16x16 pixels; block handles (b, g, tileY, tileX), all og=8
// output channels per thread. LDS: k/v halo tiles, 2*8*484*4 B = 30,976 B.
__global__ __launch_bounds__(256) void attn_tiled_kernel(
    const float* __restrict__ x,
    const float* __restrict__ wq,
    const float* __restrict__ h_emb,
    const float* __restrict__ w_emb,
    const float* __restrict__ kbuf,
    const float* __restrict__ vbuf,
    float* __restrict__ out) {
  __shared__ float k_t[OG_ * TILE_ELEMS];
  __shared__ float v_t[OG_ * TILE_ELEMS];

  const int blk = blockIdx.x;
  const int ntx = W_ / TS;                       // 6
  const int nty = H_ / TS;                       // 6
  const int tx  = blk % ntx;
  const int ty  = (blk / ntx) % nty;
  const int g   = (blk / (ntx * nty)) % G_;
  const int b   = blk / (ntx * nty * G_);
  const int tid = (int)threadIdx.x;
  const int w0  = tx * TS;
  const int h0  = ty * TS;

  const float* kg = kbuf + (size_t)b * CHW_ + (size_t)(g * OG_) * HW_;
  const float* vg = vbuf + (size_t)b * CHW_ + (size_t)(g * OG_) * HW_;

  // Cooperative staging of the 8-channel 22x22 halo tile; cells outside the
  // image are zeroed, which reproduces the reference's zero-pad exactly.
  for (int e = tid; e < OG_ * TILE_ELEMS; e += 256) {
    const int ch = e / TILE_ELEMS;
    const int re = e % TILE_ELEMS;
    const int r  = re / HALO;
    const int cc = re % HALO;
    const int gh = h0 + r - PAD_;
    const int gw = w0 + cc - PAD_;
    const bool inb = (gh >= 0) & (gh < H_) & (gw >= 0) & (gw < W_);
    const size_t gidx = (size_t)ch * HW_ + (size_t)gh * W_ + gw;
    k_t[e] = inb ? kg[gidx] : 0.0f;
    v_t[e] = inb ? vg[gidx] : 0.0f;
  }
  __syncthreads();

  const int px = tid % TS;
  const int py = tid / TS;
  const int h  = h0 + py;
  const int w  = w0 + px;

  // q for all 8 output channels of this pixel (8x8 matvec; x read once).
  float xin[IG_];
  const float* xg = x + (size_t)b * CHW_ + (size_t)(g * IG_) * HW_
                      + (size_t)h * W_ + w;
#pragma unroll
  for (int i = 0; i < IG_; ++i) xin[i] = xg[(size_t)i * HW_];

  float qv[OG_];
  const float* wqg = wq + g * OG_ * IG_;
#pragma unroll
  for (int o = 0; o < OG_; ++o) {
    float q = 0.0f;
#pragma unroll
    for (int i = 0; i < IG_; ++i) q = fmaf(wqg[o * IG_ + i], xin[i], q);
    qv[o] = q;
  }

  const int lbase = py * HALO + px;  // LDS offset of window origin (di=dj=0)
  float* outp = out + (size_t)b * CHW_ + (size_t)(g * OG_) * HW_
                    + (size_t)h * W_ + w;

  for (int o = 0; o < OG_; ++o) {
    const bool use_h = (o < OG_ / 2);
    const float* er = use_h ? (h_emb + (g * (OG_ / 2) + o) * KW_)
                            : (w_emb + (g * (OG_ / 2) + (o - OG_ / 2)) * KW_);
    float emb7[KW_];
#pragma unroll
    for (int j = 0; j < KW_; ++j) emb7[j] = er[j];

    const float* kc = &k_t[o * TILE_ELEMS + lbase];
    const float* vc = &v_t[o * TILE_ELEMS + lbase];
    const float q = qv[o];

    // Pass 1: logits (register-resident) + max. All ds offsets immediate.
    float l[KW_ * KW_];
    float m = -3.402823466e38f;
#pragma unroll
    for (int di = 0; di < KW_; ++di) {
#pragma unroll
      for (int dj = 0; dj < KW_; ++dj) {
        const float li = q * (kc[di * HALO + dj] + emb7[use_h ? di : dj]);
        l[di * KW_ + dj] = li;
        m = fmaxf(m, li);
      }
    }
    // Pass 2: exp + sum.
    float s = 0.0f;
#pragma unroll
    for (int p = 0; p < KW_ * KW_; ++p) {
      const float e = __expf(l[p] - m);
      s += e;
      l[p] = e;
    }
    // Pass 3: weighted sum of v (zeroed halo taps contribute nothing).
    float acc = 0.0f;
#pragma unroll
    for (int di = 0; di < KW_; ++di) {
#pragma unroll
      for (int dj = 0; dj < KW_; ++dj) {
        acc = fmaf(l[di * KW_ + dj], vc[di * HALO + dj], acc);
      }
    }
    outp[(size_t)o * HW_] = acc / s;
  }
}

// ---------------------------------------------------------------------------
extern "C" void kernel_launch(void* const* d_in, const int* in_sizes, int n_in,
                              void* d_out, int out_size, void* d_ws, size_t ws_size,
                              hipStream_t stream) {
  const float* x     = (const float*)d_in[0];
  const float* wq    = (const float*)d_in[1];
  const float* wk    = (const float*)d_in[2];
  const float* wv    = (const float*)d_in[3];
  const float* h_emb = (const float*)d_in[4];
  const float* w_emb = (const float*)d_in[5];
  float* out = (float*)d_out;

  // Workspace: k and v buffers, each B*C*H*W fp32 (~9.4 MB each); both are
  // fully overwritten by stage 1 every call (deterministic).
  float* kbuf = (float*)d_ws;
  float* vbuf = kbuf + (size_t)B_ * CHW_;

  // Stage 1: 4 batches * 4 group-pairs * NCHUNK waves, 1 wave per block.
  kv_proj_wmma_kernel<<<B_ * 4 * NCHUNK, 32, 0, stream>>>(x, wk, wv, kbuf, vbuf);

  // Stage 2: one block per (b, g, 16x16 tile) = 4*8*36 = 1152 blocks.
  attn_tiled_kernel<<<B_ * G_ * (H_ / TS) * (W_ / TS), 256, 0, stream>>>(
      x, wq, h_emb, w_emb, kbuf, vbuf, out);
}